// HybridQuanvolution_65481071400446
// MI455X (gfx1250) — compile-verified
//
#include <hip/hip_runtime.h>
#include <math.h>

typedef float v2f __attribute__((ext_vector_type(2)));
typedef float v8f __attribute__((ext_vector_type(8)));

// Single-wave kernel: the reference output depends on only 16 elements of x,
// the channel-0 conv taps, prototypes[:, :4], cls_w, cls_b.
// Steps:
//   1) 4 conv outputs (uniform scalar math)
//   2) 20 RBF-style kernel values, one lane each, shared via LDS
//   3) 1x20 @ 20x10 matvec as 5x V_WMMA_F32_16X16X4_F32 (fp32 tensor path)
//   4) 10-wide log_softmax, lanes 0..9 write d_out
__global__ __launch_bounds__(32) void HybridQuanvolution_kernel(
    const float* __restrict__ x,        // (32768,1,28,28)
    const float* __restrict__ conv_w,   // (4,1,2,2)
    const float* __restrict__ conv_b,   // (4,)
    const float* __restrict__ prot,     // (20,784)
    const float* __restrict__ cls_w,    // (10,20)
    const float* __restrict__ cls_b,    // (10,)
    float* __restrict__ out)            // (10,)
{
  const int lane = (int)threadIdx.x;   // 0..31, one wave32
  const int n    = lane & 15;          // WMMA column / M-row selector
  const int half = lane >> 4;          // lane half selects K pair

  // ---- Stage 1: conv taps (all uniform -> scalar loads) -------------------
  const float w00 = conv_w[0], w01 = conv_w[1], w10 = conv_w[2], w11 = conv_w[3];
  const float b0  = conv_b[0];

  float x4[4];
#pragma unroll
  for (int j = 0; j < 4; ++j) {
    const int c = 2 * j;                        // output (h=0, w=j), stride 2
    x4[j] = fmaf(w00, x[c],
            fmaf(w01, x[c + 1],
            fmaf(w10, x[28 + c],
            fmaf(w11, x[28 + c + 1], b0))));
  }

  // ---- Stage 2: kernel values k[p] = |prod_j cos((x4[j]-prot[p][j])/2)| ---
  const int p = (lane < 20) ? lane : 0;         // branch-free OOB clamp
  float prod = 1.0f;
#pragma unroll
  for (int j = 0; j < 4; ++j)
    prod *= cosf((x4[j] - prot[p * 784 + j]) * 0.5f);
  const float kv = (lane < 20) ? fabsf(prod) : 0.0f;

  __shared__ float lds_k[32];
  __shared__ float lds_l[16];
  lds_k[lane] = kv;
  __syncthreads();

  // ---- Stage 3: logits = k @ cls_w.T via V_WMMA_F32_16X16X4_F32 -----------
  // A (16x4 f32): lane L holds M=L&15; VGPR v holds K = 2*(L>=16)+v  -> only
  //               lanes with n==0 carry the k-vector (row M=0), rest zero.
  // B (4x16 f32): lane L holds N=L&15; VGPR v holds K = 2*(L>=16)+v.
  // Accumulate K=20 as 5 chunks of 4; all math stays fp32.
  v8f acc = {};
#if __has_builtin(__builtin_amdgcn_wmma_f32_16x16x4_f32)
#pragma unroll
  for (int s = 0; s < 5; ++s) {
    v2f a, b;
#pragma unroll
    for (int v = 0; v < 2; ++v) {
      const int kk = 4 * s + half * 2 + v;      // K index in [0,20)
      const float av = lds_k[kk];               // uniform-safe LDS read
      a[v] = (n == 0) ? av : 0.0f;              // row M=0 only
      const float bv = cls_w[(n < 10 ? n : 0) * 20 + kk];
      b[v] = (n < 10) ? bv : 0.0f;              // cols N>=10 padded with 0
    }
    acc = __builtin_amdgcn_wmma_f32_16x16x4_f32(
        /*neg_a=*/false, a, /*neg_b=*/false, b,
        /*c_mod=*/(short)0, acc, /*reuse_a=*/false, /*reuse_b=*/false);
  }
#else
  // Fallback (should not trigger on gfx1250): scalar matvec into acc[0]
  if (lane < 10) {
    float s = 0.0f;
    for (int q = 0; q < 20; ++q) s += lds_k[q] * cls_w[lane * 20 + q];
    acc[0] = s;
  }
#endif

  // D row M=0 lives in acc[0] of lanes 0..15 (N = lane).
  if (lane < 10) lds_l[lane] = acc[0] + cls_b[lane];
  __syncthreads();

  // ---- Stage 4: log_softmax over 10 logits --------------------------------
  if (lane < 10) {
    float m = lds_l[0];
#pragma unroll
    for (int i = 1; i < 10; ++i) m = fmaxf(m, lds_l[i]);
    float ss = 0.0f;
#pragma unroll
    for (int i = 0; i < 10; ++i) ss += expf(lds_l[i] - m);
    out[lane] = lds_l[lane] - m - logf(ss);
  }
}

extern "C" void kernel_launch(void* const* d_in, const int* in_sizes, int n_in,
                              void* d_out, int out_size, void* d_ws, size_t ws_size,
                              hipStream_t stream) {
  (void)in_sizes; (void)n_in; (void)out_size; (void)d_ws; (void)ws_size;
  HybridQuanvolution_kernel<<<1, 32, 0, stream>>>(
      (const float*)d_in[0],   // x
      (const float*)d_in[1],   // conv_w
      (const float*)d_in[2],   // conv_b
      (const float*)d_in[3],   // prototypes
      (const float*)d_in[4],   // cls_w
      (const float*)d_in[5],   // cls_b
      (float*)d_out);
}